// FourierGridModel_1726576856421
// MI455X (gfx1250) — compile-verified
//
#include <hip/hip_runtime.h>
#include <math.h>

typedef float v2f __attribute__((ext_vector_type(2)));
typedef float v8f __attribute__((ext_vector_type(8)));

#define NRAYS   16384
#define NSAMP   256
#define GRID_N  160
#define RPB     16            // rays per block: one wave32 per ray
#define SPL     8             // samples per lane: 32 lanes * 8 = 256
#define THREADS (RPB * 32)

__device__ __forceinline__ float sigmoidf(float x) {
    return 1.0f / (1.0f + __expf(-x));
}

__global__ __launch_bounds__(THREADS) void volrend_kernel(
    const float* __restrict__ ray_pts,   // [NRAYS, NSAMP, 3]
    const float* __restrict__ dgrid,     // [1, 160,160,160]
    const float* __restrict__ kgrid,     // [3, 160,160,160]
    float* __restrict__ out)             // rgb[NRAYS*3] | w[NRAYS*NSAMP] | alast[NRAYS]
{
    const int lane = threadIdx.x & 31;
    const int wave = threadIdx.x >> 5;
    const int ray  = blockIdx.x * RPB + wave;

    const float XMIN      = -1.2f;
    const float SCL       = (float)(GRID_N - 1) / 2.4f;
    const float ACT_SHIFT = -9.210240371976184f;  // log(1/(1-1e-4) - 1)
    const float INTERVAL  = 0.01f;
    const int   S1 = GRID_N, S2 = GRID_N * GRID_N, S3 = GRID_N * GRID_N * GRID_N;

    __shared__ float part[RPB][96];      // per-ray partial RGB sums, k = lane*3 + c
    __shared__ float alast_s[RPB];

    float alpha[SPL];
    float rgbv[SPL][3];

    const float* pbase = ray_pts + ((size_t)ray * NSAMP + (size_t)lane * SPL) * 3;

    #pragma unroll
    for (int i = 0; i < SPL; ++i) {
        const float px = pbase[i * 3 + 0];
        const float py = pbase[i * 3 + 1];
        const float pz = pbase[i * 3 + 2];

        float ux = fminf(fmaxf((px - XMIN) * SCL, 0.0f), (float)(GRID_N - 1));
        float uy = fminf(fmaxf((py - XMIN) * SCL, 0.0f), (float)(GRID_N - 1));
        float uz = fminf(fmaxf((pz - XMIN) * SCL, 0.0f), (float)(GRID_N - 1));
        int x0 = (int)ux, y0 = (int)uy, z0 = (int)uz;
        int x1 = min(x0 + 1, GRID_N - 1);
        int y1 = min(y0 + 1, GRID_N - 1);
        int z1 = min(z0 + 1, GRID_N - 1);
        float fx = ux - (float)x0, fy = uy - (float)y0, fz = uz - (float)z0;

        int xa = x0 * S2, xb = x1 * S2, ya = y0 * S1, yb = y1 * S1;
        int o000 = xa + ya + z0, o001 = xa + ya + z1;
        int o010 = xa + yb + z0, o011 = xa + yb + z1;
        int o100 = xb + ya + z0, o101 = xb + ya + z1;
        int o110 = xb + yb + z0, o111 = xb + yb + z1;

        // density -> alpha
        {
            float c00 = dgrid[o000] * (1.f - fz) + dgrid[o001] * fz;
            float c01 = dgrid[o010] * (1.f - fz) + dgrid[o011] * fz;
            float c10 = dgrid[o100] * (1.f - fz) + dgrid[o101] * fz;
            float c11 = dgrid[o110] * (1.f - fz) + dgrid[o111] * fz;
            float den = (c00 * (1.f - fy) + c01 * fy) * (1.f - fx)
                      + (c10 * (1.f - fy) + c11 * fy) * fx;
            float xs = den + ACT_SHIFT;
            float sp = (xs > 20.0f) ? xs : log1pf(__expf(xs));  // softplus
            alpha[i] = 1.0f - __expf(-INTERVAL * sp);           // 1-(1+e^x)^(-0.01)
        }
        // k0 3 channels -> sigmoid rgb
        #pragma unroll
        for (int c = 0; c < 3; ++c) {
            const float* g = kgrid + c * S3;
            float c00 = g[o000] * (1.f - fz) + g[o001] * fz;
            float c01 = g[o010] * (1.f - fz) + g[o011] * fz;
            float c10 = g[o100] * (1.f - fz) + g[o101] * fz;
            float c11 = g[o110] * (1.f - fz) + g[o111] * fz;
            float v = (c00 * (1.f - fy) + c01 * fy) * (1.f - fx)
                    + (c10 * (1.f - fy) + c11 * fy) * fx;
            rgbv[i][c] = sigmoidf(v);
        }
    }

    // lane-local product, then wave32 multiplicative inclusive scan
    float P = 1.0f;
    #pragma unroll
    for (int i = 0; i < SPL; ++i) P *= (1.0f - alpha[i]);
    float incl = P;
    #pragma unroll
    for (int d = 1; d < 32; d <<= 1) {
        float v = __shfl_up(incl, d, 32);
        if (lane >= d) incl *= v;
    }
    float excl = __shfl_up(incl, 1, 32);
    if (lane == 0) excl = 1.0f;
    float total = __shfl(incl, 31, 32);   // T_inc[last] = alphainv_last

    // weights + per-lane partial weighted-rgb sums
    float w[SPL];
    float ps0 = 0.f, ps1 = 0.f, ps2 = 0.f;
    float T = excl;
    #pragma unroll
    for (int i = 0; i < SPL; ++i) {
        w[i] = T * alpha[i];
        T *= (1.0f - alpha[i]);
        ps0 += w[i] * rgbv[i][0];
        ps1 += w[i] * rgbv[i][1];
        ps2 += w[i] * rgbv[i][2];
    }

    float* out_w = out + (size_t)NRAYS * 3;
    float* out_a = out + (size_t)NRAYS * 3 + (size_t)NRAYS * NSAMP;
    size_t wb = (size_t)ray * NSAMP + (size_t)lane * SPL;
    *(float4*)(out_w + wb)     = make_float4(w[0], w[1], w[2], w[3]);
    *(float4*)(out_w + wb + 4) = make_float4(w[4], w[5], w[6], w[7]);
    if (lane == 0) { out_a[ray] = total; alast_s[wave] = total; }

    part[wave][lane * 3 + 0] = ps0;
    part[wave][lane * 3 + 1] = ps1;
    part[wave][lane * 3 + 2] = ps2;
    __syncthreads();

    // Wave 0: 16x96 partials  x  96x16 selection matrix (B[k,c] = (k%3==c))
    // via 24 chained V_WMMA_F32_16X16X4_F32. Uniform branch at wave boundary
    // => EXEC all ones inside, as WMMA requires.
    if (threadIdx.x < 32) {
        const int L  = threadIdx.x;
        const int n  = L & 15;        // N column (and A row)
        const int hi = L >> 4;        // 0: K=+0,+1   1: K=+2,+3  (16x4 f32 A layout)

        // kk = 4j + 2*hi  =>  kk mod 3 = (j + 2*hi) mod 3: period 3 in j.
        // Precompute the 3 possible B operand vectors per lane so the WMMA
        // loop carries no per-iteration modulo arithmetic.
        v2f Bsel[3];
        #pragma unroll
        for (int m = 0; m < 3; ++m) {
            int km = (m + 2 * hi) % 3;
            Bsel[m].x = (km == n) ? 1.0f : 0.0f;
            Bsel[m].y = (((km + 1) % 3) == n) ? 1.0f : 0.0f;
        }

        v8f acc = {};
        #pragma unroll
        for (int j = 0; j < 24; ++j) {
            int kk = 4 * j + 2 * hi;
            v2f A;
            A.x = part[n][kk];
            A.y = part[n][kk + 1];
            acc = __builtin_amdgcn_wmma_f32_16x16x4_f32(
                /*neg_a=*/false, A, /*neg_b=*/false, Bsel[j % 3],
                /*c_mod=*/(short)0, acc, /*reuse_a=*/false, /*reuse_b=*/false);
        }
        // D layout: rows 0-7 -> lanes 0-15 (vgpr=row), rows 8-15 -> lanes 16-31
        if (n < 3) {
            const int rb = hi * 8;
            #pragma unroll
            for (int r = 0; r < 8; ++r) {
                out[(size_t)(blockIdx.x * RPB + rb + r) * 3 + n] =
                    acc[r] + alast_s[rb + r];
            }
        }
    }
}

extern "C" void kernel_launch(void* const* d_in, const int* in_sizes, int n_in,
                              void* d_out, int out_size, void* d_ws, size_t ws_size,
                              hipStream_t stream) {
    (void)in_sizes; (void)n_in; (void)d_ws; (void)ws_size; (void)out_size;
    const float* ray_pts = (const float*)d_in[0];
    const float* dgrid   = (const float*)d_in[1];
    const float* kgrid   = (const float*)d_in[2];
    float* out = (float*)d_out;

    dim3 block(THREADS);
    dim3 grid(NRAYS / RPB);   // 1024 blocks x 512 threads
    hipLaunchKernelGGL(volrend_kernel, grid, block, 0, stream,
                       ray_pts, dgrid, kgrid, out);
}